// Model_49890340110359
// MI455X (gfx1250) — compile-verified
//
#include <hip/hip_runtime.h>

// ---------------------------------------------------------------------------
// GCN 2-layer + edge MLP predictor for MI455X (gfx1250, wave32).
//
// Pipeline (all fp32, WMMA f32_16x16x4_f32 for every GEMM):
//   1) zero norms+agg
//   2) degrees via f32 atomics (exact: deg < 2^24)
//   3) norm = deg>0 ? 1/sqrt(deg) : 0   (in place)
//   4) scatter1: agg[dst] += x[src]*norm_out[src]        (102M f32 atomics, L2-resident)
//   5) gemm1:    h = relu(agg*norm_in @ W1 + b1) * norm_out   (WMMA, fused epilogue)
//   6) zero agg
//   7) scatter2: agg[dst] += h[src]                      (norm_out pre-applied in 5)
//   8) gemm2:    h = (agg*norm_in) @ W2 + b2             (WMMA)
//   9) pred:     out = [h[psrc]|h[pdst]] @ Wp + bp       (WMMA, K=128 split gather)
// ---------------------------------------------------------------------------

typedef float v2f __attribute__((ext_vector_type(2)));
typedef float v8f __attribute__((ext_vector_type(8)));

#define WMMA_F32(a, b, c) \
  __builtin_amdgcn_wmma_f32_16x16x4_f32(false, (a), false, (b), (short)0, (c), false, false)

__global__ void zero_kernel(float* __restrict__ p, long n) {
  long i = (long)blockIdx.x * blockDim.x + threadIdx.x;
  if (i < n) p[i] = 0.0f;
}

__global__ void degree_kernel(const int* __restrict__ src, const int* __restrict__ dst,
                              float* __restrict__ deg_out, float* __restrict__ deg_in, int E) {
  int e = blockIdx.x * blockDim.x + threadIdx.x;
  if (e < E) {
    atomicAdd(&deg_out[src[e]], 1.0f);
    atomicAdd(&deg_in[dst[e]], 1.0f);
  }
}

__global__ void norm_kernel(float* __restrict__ d, int n) {
  int i = blockIdx.x * blockDim.x + threadIdx.x;
  if (i < n) {
    float v = d[i];
    d[i] = (v > 0.0f) ? (1.0f / sqrtf(v)) : 0.0f;
  }
}

// One thread per (edge, feature). 64 consecutive threads share an edge so the
// 256B row gather and the 256B of scatter atomics are contiguous. Features are
// L2-resident (25.6MB << 192MB L2), so this is L2-atomic-throughput bound.
__global__ void scatter_kernel(const float* __restrict__ feat, const float* __restrict__ scale,
                               const int* __restrict__ src, const int* __restrict__ dst,
                               float* __restrict__ agg, int E) {
  int t = blockIdx.x * blockDim.x + threadIdx.x;
  if (t >= E * 64) return;
  int e = t >> 6;
  int d = t & 63;
  int s = src[e];
  float v = feat[(long)s * 64 + d];
  if (scale) v *= scale[s];
  atomicAdd(&agg[(long)dst[e] * 64 + d], v);
}

// out[N,64] = epilogue( (in * norm_in[:,None]) @ W(64x64) + bias )
// One wave per 16-row tile; 4 accumulators cover the 64 output columns.
// A layout (16x4 f32):  lane m=lane&15, kb=2*(lane>>4): v2f = in[m][k0+kb .. +1]
// B layout (4x16 f32):  lane n=lane&15, same kb:        v2f = W[k0+kb .. +1][n]
// C layout (16x16 f32): v8f[r] = out[r + 8*(lane>>4)][lane&15 (+16*nt)]
template <bool RELU, bool POST>
__global__ void gcn_gemm_kernel(const float* __restrict__ in, const float* __restrict__ norm_in,
                                const float* __restrict__ norm_post,
                                const float* __restrict__ W, const float* __restrict__ bias,
                                float* __restrict__ out, int nrows, int ntiles) {
  int wave = threadIdx.x >> 5;
  int lane = threadIdx.x & 31;
  int tile = blockIdx.x * (blockDim.x >> 5) + wave;
  if (tile >= ntiles) return;               // uniform per wave: EXEC stays full for WMMA

  int m = lane & 15;
  int half = lane >> 4;
  int kb = half * 2;
  int row = tile * 16 + m;
  if (row >= nrows) row = nrows - 1;        // clamp loads; their C rows are store-guarded
  float s = norm_in[row];
  const float* arow = in + (long)row * 64;

  v8f acc[4] = {v8f{}, v8f{}, v8f{}, v8f{}};

#pragma unroll
  for (int k0 = 0; k0 < 64; k0 += 4) {
    v2f a;
    a.x = arow[k0 + kb] * s;
    a.y = arow[k0 + kb + 1] * s;
    const float* w0 = W + (k0 + kb) * 64 + m;  // W row k0+kb, col m (+16*nt)
#pragma unroll
    for (int nt = 0; nt < 4; ++nt) {
      v2f b;
      b.x = w0[nt * 16];
      b.y = w0[64 + nt * 16];
      acc[nt] = WMMA_F32(a, b, acc[nt]);
    }
  }

#pragma unroll
  for (int nt = 0; nt < 4; ++nt) {
    int col = nt * 16 + m;
    float bb = bias[col];
#pragma unroll
    for (int r = 0; r < 8; ++r) {
      int ro = tile * 16 + r + 8 * half;
      if (ro < nrows) {
        float v = acc[nt][r] + bb;
        if (RELU) v = fmaxf(v, 0.0f);
        if (POST) v *= norm_post[ro];
        out[(long)ro * 64 + col] = v;
      }
    }
  }
}

// out[P,8] = [h[psrc] | h[pdst]] @ Wp(128x8) + bp.
// Single K=128 WMMA GEMM: k<64 gathers the psrc row, k>=64 the pdst row
// (uniform switch). N padded 8->16; cols>=8 of B zeroed via selects (no branch
// around the WMMA, EXEC all ones).
__global__ void pred_kernel(const float* __restrict__ h,
                            const int* __restrict__ psrc, const int* __restrict__ pdst,
                            const float* __restrict__ Wp, const float* __restrict__ bp,
                            float* __restrict__ out, int P, int ntiles) {
  int wave = threadIdx.x >> 5;
  int lane = threadIdx.x & 31;
  int tile = blockIdx.x * (blockDim.x >> 5) + wave;
  if (tile >= ntiles) return;

  int m = lane & 15;
  int half = lane >> 4;
  int kb = half * 2;
  int p = tile * 16 + m;
  if (p >= P) p = P - 1;
  const float* rs = h + (long)psrc[p] * 64;
  const float* rd = h + (long)pdst[p] * 64;

  v8f acc{};
#pragma unroll
  for (int k0 = 0; k0 < 128; k0 += 4) {
    const float* ar = (k0 < 64) ? rs : rd;   // uniform (k0 is compile-time)
    int ko = k0 & 63;
    v2f a;
    a.x = ar[ko + kb];
    a.y = ar[ko + kb + 1];
    // B col = m; valid cols 0..7, pad 8..15 with zeros (cndmask, no divergence)
    float w0 = Wp[(k0 + kb) * 8 + (m & 7)];
    float w1 = Wp[(k0 + kb + 1) * 8 + (m & 7)];
    v2f b;
    b.x = (m < 8) ? w0 : 0.0f;
    b.y = (m < 8) ? w1 : 0.0f;
    acc = WMMA_F32(a, b, acc);
  }

  if (m < 8) {
    float bb = bp[m];
#pragma unroll
    for (int r = 0; r < 8; ++r) {
      int po = tile * 16 + r + 8 * half;
      if (po < P) out[(long)po * 8 + m] = acc[r] + bb;
    }
  }
}

extern "C" void kernel_launch(void* const* d_in, const int* in_sizes, int n_in,
                              void* d_out, int out_size, void* d_ws, size_t ws_size,
                              hipStream_t stream) {
  const float* x   = (const float*)d_in[0];
  const int* src   = (const int*)d_in[1];
  const int* dst   = (const int*)d_in[2];
  const int* psrc  = (const int*)d_in[3];
  const int* pdst  = (const int*)d_in[4];
  const float* W1  = (const float*)d_in[5];
  const float* b1  = (const float*)d_in[6];
  const float* W2  = (const float*)d_in[7];
  const float* b2  = (const float*)d_in[8];
  const float* Wp  = (const float*)d_in[9];
  const float* bp  = (const float*)d_in[10];
  float* out = (float*)d_out;

  int N = in_sizes[0] / 64;   // 100,000
  int E = in_sizes[1];        // 1,600,000
  int P = in_sizes[3];        // 500,000

  // Workspace: norm_out[N] | norm_in[N] | agg[N*64] | h[N*64]  (~52 MB)
  float* norm_out = (float*)d_ws;
  float* norm_in  = norm_out + N;
  float* agg      = norm_in + N;
  float* h        = agg + (long)N * 64;

  long zn = 2L * N + 64L * N;  // norms + agg
  zero_kernel<<<(unsigned)((zn + 255) / 256), 256, 0, stream>>>(norm_out, zn);

  degree_kernel<<<(E + 255) / 256, 256, 0, stream>>>(src, dst, norm_out, norm_in, E);
  norm_kernel<<<(2 * N + 255) / 256, 256, 0, stream>>>(norm_out, 2 * N);

  int tE = E * 64;
  scatter_kernel<<<(tE + 255) / 256, 256, 0, stream>>>(x, norm_out, src, dst, agg, E);

  int ntiles = (N + 15) / 16;
  const int WPB = 8;  // waves per block (256 threads)
  gcn_gemm_kernel<true, true><<<(ntiles + WPB - 1) / WPB, WPB * 32, 0, stream>>>(
      agg, norm_in, norm_out, W1, b1, h, N, ntiles);

  zero_kernel<<<(unsigned)(((long)N * 64 + 255) / 256), 256, 0, stream>>>(agg, (long)N * 64);
  scatter_kernel<<<(tE + 255) / 256, 256, 0, stream>>>(h, nullptr, src, dst, agg, E);

  gcn_gemm_kernel<false, false><<<(ntiles + WPB - 1) / WPB, WPB * 32, 0, stream>>>(
      agg, norm_in, nullptr, W2, b2, h, N, ntiles);

  int ptiles = (P + 15) / 16;
  pred_kernel<<<(ptiles + WPB - 1) / WPB, WPB * 32, 0, stream>>>(
      h, psrc, pdst, Wp, bp, out, P, ptiles);
}